// PyCoxLoss_22282290332224
// MI455X (gfx1250) — compile-verified
//
#include <hip/hip_runtime.h>
#include <math.h>

// Cox partial-likelihood loss, N = 8192, CENSORING < 0 (=> yc == event).
//
// Math reduction of the reference:
//   sort by ytime ascending -> sp (risk scores in time order)
//   x_j = exp(sp_j), S_j = sum_{i>=j} x_i  (suffix sum)
//   loss = sum_j ev_j*(log(S_j) - sp_j) / sum_j ev_j
//
// Single-workgroup LDS-resident pipeline (96 KB of input => latency problem,
// not bandwidth; one WGP, 32 waves):
//   0) ytime -> LDS via GLOBAL_LOAD_ASYNC_TO_LDS_B128 (ASYNCcnt path)
//   1) bitonic sort (ytime key, u16 index payload) in LDS
//   2) x = exp(pred[idx])                             (VALU transcendental)
//   3) 16-wide segmented inclusive prefix via WMMA:   P = X * U  (upper-tri ones)
//      chained over K with 4x V_WMMA_F32_16X16X4_F32 per 16x16 tile, 1 tile/wave
//   4) 512 row totals -> Hillis-Steele scan in LDS
//   5) S_j = total - flat_prefix_incl_j + x_j ; block-reduce the loss scalar

#define N_ELEMS   8192
#define NTHREADS  1024
#define NROWS     (N_ELEMS / 16)   // 512 rows of 16 elements

typedef __attribute__((ext_vector_type(2))) float v2f;
typedef __attribute__((ext_vector_type(8))) float v8f;

// GCC-style vector to match the async-builtin prototype exactly
// (diagnostic showed: __attribute__((__vector_size__(4 * sizeof(int)))) int *)
typedef int v4i_g __attribute__((vector_size(16)));
typedef __attribute__((address_space(1))) v4i_g* g1_v4i_p;   // global
typedef __attribute__((address_space(3))) v4i_g* l3_v4i_p;   // LDS

// ---- gfx1250 async global->LDS staging (ASYNCcnt), with safe fallbacks ------
__device__ __forceinline__ void async_g2l_b128(const float* g, float* l)
{
#if __has_builtin(__builtin_amdgcn_global_load_async_to_lds_b128)
    __builtin_amdgcn_global_load_async_to_lds_b128(
        (g1_v4i_p)(const_cast<float*>(g)),
        (l3_v4i_p)(l),
        /*offset=*/0, /*cpol=*/0);
#else
    *(float4*)l = *(const float4*)g;   // plain VGPR-bounce fallback
#endif
}

__device__ __forceinline__ void wait_asynccnt0()
{
#if __has_builtin(__builtin_amdgcn_s_wait_asynccnt)
    __builtin_amdgcn_s_wait_asynccnt(0);
#else
    asm volatile("s_wait_asynccnt 0" ::: "memory");
#endif
}

__global__ __launch_bounds__(NTHREADS)
void cox_loss_kernel(const float* __restrict__ pred,
                     const float* __restrict__ ytime,
                     const int*   __restrict__ event,
                     float* __restrict__ out)
{
    // 60 KB static LDS (well under the 320 KB WGP pool)
    __shared__ __align__(16) float s_val[N_ELEMS]; // keys -> x -> row-prefix(x)
    __shared__ unsigned short s_idx[N_ELEMS];      // permutation (8192 fits u16)
    __shared__ float          s_rowA[NROWS];
    __shared__ float          s_rowB[NROWS];
    __shared__ float          s_red1[NTHREADS];
    __shared__ float          s_red2[NTHREADS];

    const int tid = threadIdx.x;

    // ---- 0) stage ytime into LDS on the async path; fill identity perm ------
    {
        const float* gsrc = ytime + tid * 8;       // 8192/1024 = 8 floats/thread
        float*       ldst = &s_val[tid * 8];       // 32B global / 16B LDS aligned
        async_g2l_b128(gsrc,     ldst);
        async_g2l_b128(gsrc + 4, ldst + 4);
    }
    __builtin_prefetch(pred + tid, 0, 0);          // warm pred for later gathers
    for (int t = tid; t < N_ELEMS; t += NTHREADS) {
        s_idx[t] = (unsigned short)t;
    }
    wait_asynccnt0();
    __syncthreads();

    // ---- 1) bitonic sort ascending on keys, carrying idx --------------------
    for (int kk = 2; kk <= N_ELEMS; kk <<= 1) {
        for (int jj = kk >> 1; jj > 0; jj >>= 1) {
            for (int t = tid; t < N_ELEMS; t += NTHREADS) {
                int ixj = t ^ jj;
                if (ixj > t) {
                    bool up  = ((t & kk) == 0);
                    float a  = s_val[t];
                    float b  = s_val[ixj];
                    if ((a > b) == up) {
                        s_val[t]   = b;  s_val[ixj] = a;
                        unsigned short ia = s_idx[t];
                        s_idx[t]   = s_idx[ixj];
                        s_idx[ixj] = ia;
                    }
                }
            }
            __syncthreads();
        }
    }

    // ---- 2) x_j = exp(sp_j) in sorted order (reuse s_val) -------------------
    for (int t = tid; t < N_ELEMS; t += NTHREADS) {
        s_val[t] = expf(pred[s_idx[t]]);
    }
    __syncthreads();

    // ---- 3) per-tile inclusive row prefix via WMMA:  P = X * U --------------
    // Wave w owns the 16x16 tile s_val[w*256 .. w*256+255] (row-major, 16 rows
    // of 16 consecutive elements). U[k][n] = (k <= n), so P[r][c] = prefix of
    // row r up to column c inclusive. K=16 is split into 4 chained K=4 WMMAs.
    //
    // VGPR layouts (ISA 7.12.2, wave32):
    //   A 16x4 f32 : lane L holds M=L&15; vgpr0 = K = (L<16 ? 0 : 2), vgpr1 = K+1
    //   B 4x16 f32 : lane L holds N=L&15; vgpr0 = K = (L<16 ? 0 : 2), vgpr1 = K+1
    //   C/D 16x16  : vgpr v -> row = v + (L<16 ? 0 : 8), col = L&15
    {
        const int wave = tid >> 5;        // 0..31, one tile each
        const int lane = tid & 31;
        const int half = lane >> 4;       // 0 or 1
        const int l16  = lane & 15;
        float* X = &s_val[wave * 256];

        v8f acc = {};                     // C starts at zero
        #pragma unroll
        for (int q = 0; q < 4; ++q) {
            const int kx = 4 * q + (half ? 2 : 0);
            const int ky = kx + 1;
            v2f a, b;
            a.x = X[l16 * 16 + kx];       // A panel = X[:, 4q:4q+4]
            a.y = X[l16 * 16 + ky];
            b.x = (kx <= l16) ? 1.0f : 0.0f;   // B panel = U[4q:4q+4, :]
            b.y = (ky <= l16) ? 1.0f : 0.0f;
            acc = __builtin_amdgcn_wmma_f32_16x16x4_f32(
                      /*neg_a=*/false, a, /*neg_b=*/false, b,
                      /*c_mod=*/(short)0, acc,
                      /*reuse_a=*/false, /*reuse_b=*/false);
        }
        // In-place writeback: all LDS reads of this wave precede these stores
        // in program order (same-wave DS ops stay ordered).
        #pragma unroll
        for (int v = 0; v < 8; ++v) {
            X[(v + 8 * half) * 16 + l16] = acc[v];
        }
    }
    __syncthreads();

    // ---- 4) inclusive scan of the 512 row totals ----------------------------
    float* src = s_rowA;
    float* dst = s_rowB;
    if (tid < NROWS) s_rowA[tid] = s_val[tid * 16 + 15];   // rowTotal[r]
    __syncthreads();
    for (int off = 1; off < NROWS; off <<= 1) {
        if (tid < NROWS) {
            float v = src[tid];
            if (tid >= off) v += src[tid - off];
            dst[tid] = v;
        }
        __syncthreads();
        float* tmp = src; src = dst; dst = tmp;
    }
    const float* rowIncl = src;           // inclusive scan of row totals
    const float  total   = rowIncl[NROWS - 1];

    // ---- 5) per-element terms + block reduction -----------------------------
    float termSum = 0.0f, cntSum = 0.0f;
    for (int t = tid; t < N_ELEMS; t += NTHREADS) {
        const int   g  = s_idx[t];
        const float sp = pred[g];
        const float ev = (float)event[g];           // CENSORING<0 -> yc = event
        const int   r  = t >> 4;
        const float rowExcl = (r == 0) ? 0.0f : rowIncl[r - 1];
        const float pin = s_val[t] + rowExcl;       // inclusive flat prefix of x
        const float x   = expf(sp);
        const float S   = total - pin + x;          // suffix sum incl. self
        termSum += ev * (logf(S) - sp);
        cntSum  += ev;
    }
    s_red1[tid] = termSum;
    s_red2[tid] = cntSum;
    __syncthreads();
    for (int s = NTHREADS / 2; s > 0; s >>= 1) {
        if (tid < s) {
            s_red1[tid] += s_red1[tid + s];
            s_red2[tid] += s_red2[tid + s];
        }
        __syncthreads();
    }
    if (tid == 0) out[0] = s_red1[0] / s_red2[0];
}

extern "C" void kernel_launch(void* const* d_in, const int* in_sizes, int n_in,
                              void* d_out, int out_size, void* d_ws, size_t ws_size,
                              hipStream_t stream)
{
    (void)in_sizes; (void)n_in; (void)out_size; (void)d_ws; (void)ws_size;
    const float* pred  = (const float*)d_in[0];
    const float* ytime = (const float*)d_in[1];
    const int*   event = (const int*)  d_in[2];
    float*       out   = (float*)d_out;

    cox_loss_kernel<<<1, NTHREADS, 0, stream>>>(pred, ytime, event, out);
}